// IzhikevichNet_39633958207940
// MI455X (gfx1250) — compile-verified
//
#include <hip/hip_runtime.h>
#include <math.h>

// ---- problem sizes (from the reference) ------------------------------------
#define HID    8192
#define INP    4096
#define OUTN   4096
#define NTOT   (HID + OUTN)   // 12288
#define STEPS  100
#define NSLICE 8              // split-K factor (for memory-level parallelism)

typedef float v2f __attribute__((ext_vector_type(2)));
typedef float v8f __attribute__((ext_vector_type(8)));

// Zero the loss accumulator (d_out[0]) — must run every launch since the
// harness does not re-poison between graph replays.
__global__ void zero_loss_kernel(float* loss) {
    if (threadIdx.x == 0 && blockIdx.x == 0) loss[0] = 0.0f;
}

// ---------------------------------------------------------------------------
// Split-K partial GEMV via V_WMMA_F32_16X16X4_F32.
// grid.x * wavesPerBlock covers 16-row tiles, grid.y = K-slice.
// One wave = one (16-row tile, K-slice) pair. Lanes 0-15 feed A's K={0,1},
// lanes 16-31 feed K={2,3} (per the 32-bit 16x4 A layout), so each lane loads
// float4s of its row with lanes>=16 shifted +4 columns; B broadcasts x[k]
// across all 16 N columns, making every accumulator column equal to the
// partial dot product. Lane 0 ends with rows mbase..+7 in its 8 C VGPRs,
// lane 16 with rows mbase+8..+15. Partials are written per-slice (no atomics
// => deterministic); a combine kernel sums slices and applies compress().
// ---------------------------------------------------------------------------
__global__ __launch_bounds__(256)
void gemv_partial_wmma(const float* __restrict__ W,
                       const float* __restrict__ x,
                       float* __restrict__ partial,   // [NSLICE][rows]
                       int K, int rows)
{
    const int lane   = threadIdx.x & 31;
    const int wave   = threadIdx.x >> 5;
    const int tile   = blockIdx.x * (blockDim.x >> 5) + wave;  // 16-row tile
    const int slice  = blockIdx.y;
    const int kslice = K / NSLICE;
    const int kbeg   = slice * kslice;

    const int mbase = tile << 4;
    const int row   = mbase + (lane & 15);
    const int koff  = (lane >> 4) << 2;                        // 0 or 4

    const float* wrow = W + (size_t)row * (size_t)K + kbeg + koff;
    const float* xoff = x + kbeg + koff;

    v8f c = {0.f, 0.f, 0.f, 0.f, 0.f, 0.f, 0.f, 0.f};

    for (int k = 0; k < kslice; k += 16) {
        __builtin_prefetch(wrow + k + 256, 0, 3);              // global_prefetch_b8
        float4 w0 = *(const float4*)(wrow + k);
        float4 w1 = *(const float4*)(wrow + k + 8);
        float4 x0 = *(const float4*)(xoff + k);
        float4 x1 = *(const float4*)(xoff + k + 8);

        v2f a, b;
        a[0] = w0.x; a[1] = w0.y;  b[0] = x0.x; b[1] = x0.y;
        c = __builtin_amdgcn_wmma_f32_16x16x4_f32(false, a, false, b, (short)0, c, false, false);
        a[0] = w0.z; a[1] = w0.w;  b[0] = x0.z; b[1] = x0.w;
        c = __builtin_amdgcn_wmma_f32_16x16x4_f32(false, a, false, b, (short)0, c, false, false);
        a[0] = w1.x; a[1] = w1.y;  b[0] = x1.x; b[1] = x1.y;
        c = __builtin_amdgcn_wmma_f32_16x16x4_f32(false, a, false, b, (short)0, c, false, false);
        a[0] = w1.z; a[1] = w1.w;  b[0] = x1.z; b[1] = x1.w;
        c = __builtin_amdgcn_wmma_f32_16x16x4_f32(false, a, false, b, (short)0, c, false, false);
    }

    if (lane == 0 || lane == 16) {
        const int base = mbase + ((lane >> 4) << 3);
        float* dst = partial + (size_t)slice * rows + base;
#pragma unroll
        for (int g = 0; g < 8; ++g) dst[g] = c[g];
    }
}

// Sum the NSLICE partial strips, apply compress(); optionally accumulate MSE.
__global__ __launch_bounds__(256)
void combine_compress(const float* __restrict__ partial,   // [NSLICE][rows]
                      float* __restrict__ y,
                      int rows,
                      const float* __restrict__ gain_p,
                      const float* __restrict__ target,    // null => no loss
                      float* __restrict__ loss_accum)
{
    const int i = blockIdx.x * blockDim.x + threadIdx.x;
    if (i >= rows) return;
    float s = 0.f;
#pragma unroll
    for (int sl = 0; sl < NSLICE; ++sl) s += partial[(size_t)sl * rows + i];
    const float gain = gain_p[0];
    float comp = copysignf(log1pf(fmaf(gain, fabsf(s), 1e-6f)), s);
    y[i] = comp;
    if (target) {
        float d = comp - target[i];
        atomicAdd(loss_accum, d * d * (1.0f / 4096.0f));
    }
}

// ---------------------------------------------------------------------------
// Fallback: fused single-pass GEMV (used only if workspace is tiny).
// ---------------------------------------------------------------------------
__global__ __launch_bounds__(256)
void gemv_compress_wmma(const float* __restrict__ W,
                        const float* __restrict__ x,
                        float* __restrict__ y,
                        int K,
                        const float* __restrict__ gain_p,
                        const float* __restrict__ target,
                        float* __restrict__ loss_accum)
{
    const int lane  = threadIdx.x & 31;
    const int wave  = threadIdx.x >> 5;
    const int tile  = blockIdx.x * (blockDim.x >> 5) + wave;
    const int mbase = tile << 4;
    const int row   = mbase + (lane & 15);
    const int koff  = (lane >> 4) << 2;
    const float gain = gain_p[0];

    const float* wrow = W + (size_t)row * (size_t)K + koff;
    const float* xoff = x + koff;

    v8f c = {0.f, 0.f, 0.f, 0.f, 0.f, 0.f, 0.f, 0.f};
    for (int k = 0; k < K; k += 16) {
        __builtin_prefetch(wrow + k + 256, 0, 3);
        float4 w0 = *(const float4*)(wrow + k);
        float4 w1 = *(const float4*)(wrow + k + 8);
        float4 x0 = *(const float4*)(xoff + k);
        float4 x1 = *(const float4*)(xoff + k + 8);
        v2f a, b;
        a[0] = w0.x; a[1] = w0.y;  b[0] = x0.x; b[1] = x0.y;
        c = __builtin_amdgcn_wmma_f32_16x16x4_f32(false, a, false, b, (short)0, c, false, false);
        a[0] = w0.z; a[1] = w0.w;  b[0] = x0.z; b[1] = x0.w;
        c = __builtin_amdgcn_wmma_f32_16x16x4_f32(false, a, false, b, (short)0, c, false, false);
        a[0] = w1.x; a[1] = w1.y;  b[0] = x1.x; b[1] = x1.y;
        c = __builtin_amdgcn_wmma_f32_16x16x4_f32(false, a, false, b, (short)0, c, false, false);
        a[0] = w1.z; a[1] = w1.w;  b[0] = x1.z; b[1] = x1.w;
        c = __builtin_amdgcn_wmma_f32_16x16x4_f32(false, a, false, b, (short)0, c, false, false);
    }
    if (lane == 0 || lane == 16) {
        const int base = mbase + ((lane >> 4) << 3);
        float lsum = 0.f;
#pragma unroll
        for (int g = 0; g < 8; ++g) {
            float val  = c[g];
            float comp = copysignf(log1pf(fmaf(gain, fabsf(val), 1e-6f)), val);
            y[base + g] = comp;
            if (target) { float d = comp - target[base + g]; lsum = fmaf(d, d, lsum); }
        }
        if (target) atomicAdd(loss_accum, lsum * (1.0f / 4096.0f));
    }
}

// ---------------------------------------------------------------------------
// Izhikevich dynamics: one thread per neuron, 100 steps, record v/u per step.
// ---------------------------------------------------------------------------
__global__ __launch_bounds__(256)
void izhikevich_dynamics(const float* __restrict__ total_input,
                         float* __restrict__ v_out,
                         float* __restrict__ u_out)
{
    const int i = blockIdx.x * blockDim.x + threadIdx.x;
    if (i >= NTOT) return;

    const float I = total_input[i];
    float v = -65.0f;
    float u = 0.2f * v;            // b * v0 = -13
    v_out[i] = v;
    u_out[i] = u;

    for (int t = 1; t < STEPS; ++t) {
        float vn = v + (0.04f * v * v + 5.0f * v + 0.14f - u + I);
        float un = u + 0.02f * (0.2f * v - u);
        bool sp = (vn >= 30.0f);
        v = sp ? -65.0f : vn;
        u = sp ? (un + 8.0f) : un;
        v_out[(size_t)t * NTOT + i] = v;
        u_out[(size_t)t * NTOT + i] = u;
    }
}

// ---------------------------------------------------------------------------
// inputs (setup_inputs order): v_input[4096], target_output[4096],
// W1[8192*4096], W2[4096*8192], gain1[1], gain2[1]
// output: [loss(1), v_vec(100*12288), u_vec(100*12288)]  (float32)
// ---------------------------------------------------------------------------
extern "C" void kernel_launch(void* const* d_in, const int* in_sizes, int n_in,
                              void* d_out, int out_size, void* d_ws, size_t ws_size,
                              hipStream_t stream)
{
    const float* v_input = (const float*)d_in[0];
    const float* target  = (const float*)d_in[1];
    const float* W1      = (const float*)d_in[2];
    const float* W2      = (const float*)d_in[3];
    const float* gain1   = (const float*)d_in[4];
    const float* gain2   = (const float*)d_in[5];

    float* out = (float*)d_out;

    // Workspace layout (split-K path):
    //   partial1 [NSLICE][HID]   partial2 [NSLICE][OUTN]   I_hidden[HID] I_output[OUTN]
    const size_t needSplit = (size_t)(NSLICE * HID + NSLICE * OUTN + NTOT) * sizeof(float);

    zero_loss_kernel<<<1, 32, 0, stream>>>(out);

    if (ws_size >= needSplit) {
        float* partial1 = (float*)d_ws;
        float* partial2 = partial1 + (size_t)NSLICE * HID;
        float* I_hidden = partial2 + (size_t)NSLICE * OUTN;
        float* I_output = I_hidden + HID;     // contiguous => total_input

        // GEMV1: 512 tiles x 8 slices = 4096 waves (~20 MB reads in flight)
        dim3 g1(HID / 16 / 8, NSLICE);
        gemv_partial_wmma<<<g1, 256, 0, stream>>>(W1, v_input, partial1, INP, HID);
        combine_compress<<<HID / 256, 256, 0, stream>>>(
            partial1, I_hidden, HID, gain1, nullptr, nullptr);

        // GEMV2: 256 tiles x 8 slices = 2048 waves
        dim3 g2(OUTN / 16 / 8, NSLICE);
        gemv_partial_wmma<<<g2, 256, 0, stream>>>(W2, I_hidden, partial2, HID, OUTN);
        combine_compress<<<OUTN / 256, 256, 0, stream>>>(
            partial2, I_output, OUTN, gain2, target, out);

        izhikevich_dynamics<<<(NTOT + 255) / 256, 256, 0, stream>>>(
            I_hidden, out + 1, out + 1 + (size_t)STEPS * NTOT);
    } else {
        // Fallback: fused single-pass (needs only 48 KB of workspace)
        float* I_hidden = (float*)d_ws;
        float* I_output = I_hidden + HID;
        gemv_compress_wmma<<<HID / 16 / 8, 256, 0, stream>>>(
            W1, v_input, I_hidden, INP, gain1, nullptr, nullptr);
        gemv_compress_wmma<<<OUTN / 16 / 8, 256, 0, stream>>>(
            W2, I_hidden, I_output, HID, gain2, target, out);
        izhikevich_dynamics<<<(NTOT + 255) / 256, 256, 0, stream>>>(
            I_hidden, out + 1, out + 1 + (size_t)STEPS * NTOT);
    }
}